// NoisyTopKRouter_23398981829023
// MI455X (gfx1250) — compile-verified
//
#include <hip/hip_runtime.h>
#include <hip/hip_bf16.h>
#include <math.h>

// ---------------- problem constants ----------------
#define N_TOK   4096
#define D_MODEL 1024
#define N_EXP   16
#define TOPK    2
#define CAP     640      // ceil(1.25 * 4096*2 / 16)

// output layout (floats), tuple flattened in return order
#define OFF_DISP   0ul
#define OFF_COMB   41943040ul                 // 4096*16*640
#define OFF_IDX    83886080ul                 // 2*41943040
#define OFF_GATES  83894272ul                 // +8192
#define OFF_BAL    83902464ul                 // +8192
#define OFF_Z      83902465ul
#define OFF_IMP    83902466ul
#define OFF_LOAD   83902482ul

// ws layout (float units)
#define WS_LOGITS  0ul        // 65536 floats
#define WS_E0      65536ul    // 4096 ints
#define WS_E1      69632ul
#define WS_G0      73728ul    // 4096 floats
#define WS_G1      77824ul
#define WS_P0      81920ul    // 4096 ints
#define WS_P1      86016ul
#define WS_IMP     90112ul    // 16 floats (accumulator)
#define WS_ZSUM    90128ul    // 1 float  (accumulator)

typedef __attribute__((ext_vector_type(2))) float v2f;
typedef __attribute__((ext_vector_type(4))) float v4f;
typedef __attribute__((ext_vector_type(8))) float v8f;

// ---------------- kernel 1: zero-fill big tensors (NT streaming stores) + accumulators ----------
// 336 MB > 192 MB L2: use non-temporal stores so the fill streams to HBM
// without evicting the whole L2 every call.
__global__ __launch_bounds__(256) void fill_kernel(float* __restrict__ out,
                                                   float* __restrict__ imp_g,
                                                   float* __restrict__ zsum_g) {
    const size_t stride = (size_t)gridDim.x * blockDim.x;
    size_t i = (size_t)blockIdx.x * blockDim.x + threadIdx.x;
    v4f z4 = {0.f, 0.f, 0.f, 0.f};
    v4f* p = (v4f*)out;
    // dispatch + combine = 83,886,080 floats = 20,971,520 float4s
    for (size_t idx = i; idx < 20971520ul; idx += stride)
        __builtin_nontemporal_store(z4, p + idx);
    if (blockIdx.x == 0) {
        if (threadIdx.x < N_EXP) imp_g[threadIdx.x] = 0.f;
        if (threadIdx.x == N_EXP) zsum_g[0] = 0.f;
    }
}

// ---------------- kernel 2: logits = tokens @ W^T + b  (V_WMMA_F32_16X16X4_F32) ----------------
// One wave per 16-row tile of tokens; full E=16 output width per tile.
// A 16x4 layout: lane l<16 holds row l, K={kk,kk+1}; lane>=16 holds row l-16, K={kk+2,kk+3}.
// B 4x16 layout: lane n<16 holds col n, K={kk,kk+1}; lanes>=16 K={kk+2,kk+3}.  B[k][n] = W[n][k].
// 4 independent accumulators break the WMMA RAW chain (only ~1 wave/SIMD at this grid size).
__global__ __launch_bounds__(32) void gemm_logits_kernel(const float* __restrict__ tokens,
                                                         const float* __restrict__ W,
                                                         const float* __restrict__ bias,
                                                         float* __restrict__ logits) {
    const int lane = threadIdx.x;          // 0..31
    const int m0   = blockIdx.x * 16;
    const int r    = lane & 15;            // row (A) / col (B)
    const int koff = (lane >> 4) * 2;      // 0 or 2
    const float* __restrict__ arow = tokens + (size_t)(m0 + r) * D_MODEL + koff;
    const float* __restrict__ brow = W      + (size_t)r        * D_MODEL + koff;

    v8f acc0 = {}, acc1 = {}, acc2 = {}, acc3 = {};
    for (int kk = 0; kk < D_MODEL; kk += 16) {
        v2f a0 = *(const v2f*)(arow + kk +  0);
        v2f b0 = *(const v2f*)(brow + kk +  0);
        v2f a1 = *(const v2f*)(arow + kk +  4);
        v2f b1 = *(const v2f*)(brow + kk +  4);
        v2f a2 = *(const v2f*)(arow + kk +  8);
        v2f b2 = *(const v2f*)(brow + kk +  8);
        v2f a3 = *(const v2f*)(arow + kk + 12);
        v2f b3 = *(const v2f*)(brow + kk + 12);
        acc0 = __builtin_amdgcn_wmma_f32_16x16x4_f32(false, a0, false, b0, (short)0, acc0, false, false);
        acc1 = __builtin_amdgcn_wmma_f32_16x16x4_f32(false, a1, false, b1, (short)0, acc1, false, false);
        acc2 = __builtin_amdgcn_wmma_f32_16x16x4_f32(false, a2, false, b2, (short)0, acc2, false, false);
        acc3 = __builtin_amdgcn_wmma_f32_16x16x4_f32(false, a3, false, b3, (short)0, acc3, false, false);
    }
    v8f acc = (acc0 + acc1) + (acc2 + acc3);

    const float bv = bias[r];
    const int rowbase = m0 + 8 * (lane >> 4);   // C/D: VGPR j -> M = j + 8*(lane/16), N = lane&15
    #pragma unroll
    for (int j = 0; j < 8; ++j)
        logits[(size_t)(rowbase + j) * N_EXP + r] = acc[j] + bv;
}

// ---------------- kernel 3: softmax / top-2 / gates / importance / z-loss partials ----------------
__global__ __launch_bounds__(256) void router_kernel(const float* __restrict__ logits,
                                                     float* __restrict__ out,
                                                     int* __restrict__ e0, int* __restrict__ e1,
                                                     float* __restrict__ g0, float* __restrict__ g1,
                                                     float* __restrict__ imp_g,
                                                     float* __restrict__ zsum_g) {
    __shared__ float simp[N_EXP];
    __shared__ float szs;
    const int tid = threadIdx.x;
    if (tid < N_EXP) simp[tid] = 0.f;
    if (tid == 0) szs = 0.f;
    __syncthreads();

    const int t = blockIdx.x * 256 + tid;
    float l[N_EXP];
    const float4* lp = (const float4*)(logits + (size_t)t * N_EXP);
    #pragma unroll
    for (int i = 0; i < 4; ++i) {
        float4 v = lp[i];
        l[4*i+0] = v.x; l[4*i+1] = v.y; l[4*i+2] = v.z; l[4*i+3] = v.w;
    }

    float m = l[0];
    #pragma unroll
    for (int i = 1; i < N_EXP; ++i) m = fmaxf(m, l[i]);

    float s = 0.f, z = 0.f, p[N_EXP];
    #pragma unroll
    for (int i = 0; i < N_EXP; ++i) { p[i] = expf(l[i] - m); s += p[i]; z += l[i] * l[i]; }
    const float inv = 1.f / s;
    #pragma unroll
    for (int i = 0; i < N_EXP; ++i) atomicAdd(&simp[i], p[i] * inv);
    atomicAdd(&szs, z);

    // top-2 (ties -> lowest index, matching lax.top_k)
    int i0 = 0; float v0 = l[0];
    #pragma unroll
    for (int i = 1; i < N_EXP; ++i) if (l[i] > v0) { v0 = l[i]; i0 = i; }
    int i1 = (i0 == 0) ? 1 : 0; float v1 = l[i1];
    #pragma unroll
    for (int i = 0; i < N_EXP; ++i) if (i != i0 && l[i] > v1) { v1 = l[i]; i1 = i; }

    const float ex = expf(v1 - v0);
    const float ga = 1.f / (1.f + ex);
    const float gb = ex / (1.f + ex);

    out[OFF_IDX   + 2*t + 0] = (float)i0;
    out[OFF_IDX   + 2*t + 1] = (float)i1;
    out[OFF_GATES + 2*t + 0] = ga;
    out[OFF_GATES + 2*t + 1] = gb;
    e0[t] = i0; e1[t] = i1; g0[t] = ga; g1[t] = gb;

    __syncthreads();
    if (tid < N_EXP) atomicAdd(&imp_g[tid], simp[tid]);
    if (tid == 0)    atomicAdd(zsum_g, szs);
}

// ---------------- kernel 4: faithful serial rank/capacity scan (1 wave, 1 thread per expert) ------
__global__ __launch_bounds__(32) void scan_kernel(const int* __restrict__ e0,
                                                  const int* __restrict__ e1,
                                                  int* __restrict__ pos0, int* __restrict__ pos1,
                                                  const float* __restrict__ imp_g,
                                                  const float* __restrict__ zsum_g,
                                                  float* __restrict__ out) {
    __shared__ int se0[N_TOK];
    __shared__ int se1[N_TOK];
    __shared__ int sload[N_EXP];
    const int tid = threadIdx.x;

    // stage expert ids into LDS (16 KB + 16 KB)
    const int4* s0 = (const int4*)e0;
    const int4* s1 = (const int4*)e1;
    int4* d0 = (int4*)se0;
    int4* d1 = (int4*)se1;
    for (int i = tid; i < N_TOK / 4; i += 32) { d0[i] = s0[i]; d1[i] = s1[i]; }
    __syncthreads();

    if (tid < N_EXP) {
        const int me = tid;
        int cnt = 0;
        for (int t = 0; t < N_TOK; ++t)
            if (se0[t] == me) { int pp = cnt++; pos0[t] = (pp < CAP) ? pp : -1; }
        const int off  = (cnt < CAP) ? cnt : CAP;     // offset = min(count_p0, C)
        int kept = off;                               // kept at p=0
        int cnt1 = 0;
        for (int t = 0; t < N_TOK; ++t)
            if (se1[t] == me) {
                int pp = off + cnt1++;
                if (pp < CAP) { pos1[t] = pp; ++kept; } else pos1[t] = -1;
            }
        sload[me] = kept;
        out[OFF_LOAD + me] = (float)kept;
        out[OFF_IMP  + me] = imp_g[me];
    }
    __syncthreads();

    if (tid == 0) {
        float iv[N_EXP];
        float mi = 0.f, ml = 0.f;
        #pragma unroll
        for (int i = 0; i < N_EXP; ++i) { iv[i] = imp_g[i]; mi += iv[i]; ml += (float)sload[i]; }
        mi *= (1.f / N_EXP); ml *= (1.f / N_EXP);
        float vi = 0.f, vl = 0.f;
        #pragma unroll
        for (int i = 0; i < N_EXP; ++i) {
            float di = iv[i] - mi;            vi += di * di;
            float dl = (float)sload[i] - ml;  vl += dl * dl;
        }
        vi *= (1.f / N_EXP); vl *= (1.f / N_EXP);
        const float eps = 1e-9f;
        const float cvi = vi / (fmaxf(mi, eps) * fmaxf(mi, eps) + eps);
        const float cvl = vl / (fmaxf(ml, eps) * fmaxf(ml, eps) + eps);
        out[OFF_BAL] = 0.5f * (cvi + cvl);
        out[OFF_Z]   = zsum_g[0] * (1.f / (float)(N_TOK * N_EXP)) * 1e-4f;
    }
}

// ---------------- kernel 5: sparse scatter into dispatch/combine ----------------
__global__ __launch_bounds__(256) void scatter_kernel(const int* __restrict__ e0,
                                                      const int* __restrict__ e1,
                                                      const float* __restrict__ g0,
                                                      const float* __restrict__ g1,
                                                      const int* __restrict__ pos0,
                                                      const int* __restrict__ pos1,
                                                      float* __restrict__ out) {
    const int t = blockIdx.x * 256 + threadIdx.x;
    if (t >= N_TOK) return;
    const int   p0 = pos0[t], p1 = pos1[t];
    const float a  = g0[t],   b  = g1[t];
    const float ka = (p0 >= 0) ? a : 0.f;
    const float kb = (p1 >= 0) ? b : 0.f;
    const float denom = fmaxf(ka + kb, 1e-8f);
    float* __restrict__ disp = out + OFF_DISP;
    float* __restrict__ comb = out + OFF_COMB;
    if (p0 >= 0) {
        size_t idx = (size_t)t * (N_EXP * CAP) + (size_t)e0[t] * CAP + p0;
        disp[idx] = 1.f;
        comb[idx] = a / denom;
    }
    if (p1 >= 0) {
        size_t idx = (size_t)t * (N_EXP * CAP) + (size_t)e1[t] * CAP + p1;
        disp[idx] = 1.f;
        comb[idx] = b / denom;
    }
}

// ---------------- launch ----------------
extern "C" void kernel_launch(void* const* d_in, const int* in_sizes, int n_in,
                              void* d_out, int out_size, void* d_ws, size_t ws_size,
                              hipStream_t stream) {
    const float* tokens = (const float*)d_in[0];   // (4096,1024)
    const float* W      = (const float*)d_in[1];   // (16,1024)
    const float* bias   = (const float*)d_in[2];   // (16,)
    // d_in[3] = noise_scale (zeros, unused by reference math)
    float* out = (float*)d_out;

    float* ws      = (float*)d_ws;
    float* logits  = ws + WS_LOGITS;
    int*   e0      = (int*)(ws + WS_E0);
    int*   e1      = (int*)(ws + WS_E1);
    float* g0      = ws + WS_G0;
    float* g1      = ws + WS_G1;
    int*   pos0    = (int*)(ws + WS_P0);
    int*   pos1    = (int*)(ws + WS_P1);
    float* imp_g   = ws + WS_IMP;
    float* zsum_g  = ws + WS_ZSUM;

    // 1) zero dispatch/combine (bandwidth floor: 336 MB of NT streaming stores) + accumulators
    fill_kernel<<<4096, 256, 0, stream>>>(out, imp_g, zsum_g);
    // 2) logits via V_WMMA_F32_16X16X4_F32 (one wave per 16-row tile, 4 independent accumulators)
    gemm_logits_kernel<<<N_TOK / 16, 32, 0, stream>>>(tokens, W, bias, logits);
    // 3) per-token softmax / top-2 / gates; importance + z-loss accumulation
    router_kernel<<<N_TOK / 256, 256, 0, stream>>>(logits, out, e0, e1, g0, g1, imp_g, zsum_g);
    // 4) faithful sequential rank/capacity assignment + losses
    scan_kernel<<<1, 32, 0, stream>>>(e0, e1, pos0, pos1, imp_g, zsum_g, out);
    // 5) scatter <=4 nonzeros per token
    scatter_kernel<<<N_TOK / 256, 256, 0, stream>>>(e0, e1, g0, g1, pos0, pos1, out);
}